// KGCL_33079838114571
// MI455X (gfx1250) — compile-verified
//
#include <hip/hip_runtime.h>
#include <hip/hip_bf16.h>

// ---------------------------------------------------------------------------
// KGCL: GAT item aggregation (WMMA f32 16x16x4 + async-to-LDS entity gather)
//       + LightGCN 3-layer SpMM + batched dot product. fp32 throughout.
// ---------------------------------------------------------------------------

typedef __attribute__((ext_vector_type(2))) float v2f;
typedef __attribute__((ext_vector_type(8))) float v8f;

#define D_EMB     64
#define K_NBR     16
#define TWO_D     128
#define A_STRIDE  132          // padded a_input row stride (floats)
#define WP_STRIDE 160          // padded K-pair row stride (floats); 160 % 64 == 32
#define WAVES_PB  2            // items (waves) per block in GAT kernel
#define PAD_ENT   80000        // padding entity id == E_ENT
#define NEG_INF_F (-9e15f)

// LDS byte offset of a generic pointer to __shared__ (low 32 bits of aperture addr)
__device__ __forceinline__ unsigned lds_off(const void* p) {
    return (unsigned)(uintptr_t)p;
}

// CDNA5 async DMA: each lane copies 8 bytes from the global address in gaddr to the LDS address in lds. Tracked by ASYNCcnt.
__device__ __forceinline__ void async_ld_b64(unsigned lds, const void* gaddr) {
    asm volatile("global_load_async_to_lds_b64 %0, %1, off"
                 :: "v"(lds), "v"(gaddr) : "memory");
}
__device__ __forceinline__ void wait_async0() {
    asm volatile("s_wait_asynccnt 0" ::: "memory");
}

// ---------------------------------------------------------------------------
// Kernel 1: GAT over KG neighbors. One wave == one item.
//   D[16x64] = A[16x128] x fc_W[128x64]  via V_WMMA_F32_16X16X4_F32
//   then e = leakyrelu(sum_n (D+b)*rel), masked softmax over K=16,
//   items_kg = att @ ent + item_emb   -> written into all_emb[U+i]
// ---------------------------------------------------------------------------
__global__ __launch_bounds__(WAVES_PB * 32)
void kgcl_gat_wmma(const float* __restrict__ item_emb,
                   const float* __restrict__ entity_emb,
                   const float* __restrict__ relation_emb,
                   const int*   __restrict__ kg_entities,
                   const int*   __restrict__ kg_relations,
                   const float* __restrict__ fc_W,
                   const float* __restrict__ fc_b,
                   float*       __restrict__ out_items)   // &all_emb[U*64]
{
    // fc_W swizzled by K-pairs: s_Wp[p*WP_STRIDE + 2n] = {W[2p][n], W[2p+1][n]}
    __shared__ __align__(16) float s_Wp[(TWO_D / 2) * WP_STRIDE];       // 40 KB
    __shared__ __align__(16) float s_a[WAVES_PB][K_NBR * A_STRIDE];     // 16.5 KB
    __shared__ float s_att[WAVES_PB][K_NBR];
    __shared__ float s_b[D_EMB];

    const int tid  = threadIdx.x;
    const int wave = tid >> 5;
    const int lane = tid & 31;
    const int item = blockIdx.x * WAVES_PB + wave;        // I divisible by WAVES_PB

    // --- async DMA the 16 scattered entity rows into the a_input tile -------
    // one GLOBAL_LOAD_ASYNC_TO_LDS_B64 per row: 32 lanes x 8B = 256B row
    const float* irow = item_emb + (size_t)item * D_EMB;
    for (int k = 0; k < K_NBR; ++k) {
        const int ek = kg_entities[item * K_NBR + k];
        const float* erow = entity_emb + (size_t)ek * D_EMB;
        async_ld_b64(lds_off(&s_a[wave][k * A_STRIDE + D_EMB]) + lane * 8,
                     erow + lane * 2);
    }

    // --- meanwhile stage fc_W (pair-swizzled) + fc_b with normal loads ------
    for (int idx = tid; idx < (TWO_D / 2) * D_EMB; idx += WAVES_PB * 32) {
        const int p = idx >> 6;           // K-pair
        const int n = idx & 63;           // column
        s_Wp[p * WP_STRIDE + 2 * n]     = fc_W[(2 * p)     * D_EMB + n];
        s_Wp[p * WP_STRIDE + 2 * n + 1] = fc_W[(2 * p + 1) * D_EMB + n];
    }
    if (tid < D_EMB) s_b[tid] = fc_b[tid];

    // --- broadcast item row into cols [0,64) of every a_input row ----------
    const float i0 = irow[lane], i1 = irow[lane + 32];
    for (int k = 0; k < K_NBR; ++k) {
        s_a[wave][k * A_STRIDE + lane]      = i0;
        s_a[wave][k * A_STRIDE + lane + 32] = i1;
    }

    wait_async0();        // drain this wave's async DMAs (ds stores drain at barrier)
    __syncthreads();

    // ---- WMMA GEMM: 4 N-tiles of 16, K loop 128 in steps of 4 ----
    // A 16x4 f32 layout: lanes0-15 M=lane {K=k0,k0+1}, lanes16-31 M=lane-16 {K=k0+2,k0+3}
    // B 4x16 f32 layout mirrored: lanes0-15 rows {k0,k0+1}, lanes16-31 rows {k0+2,k0+3}
    v8f acc[4] = {};
    const int m  = lane & 15;
    const int kq = (lane >> 4) * 2;         // 0 or 2
    for (int k0 = 0; k0 < TWO_D; k0 += 4) {
        const v2f a = *(const v2f*)&s_a[wave][m * A_STRIDE + k0 + kq];
        const int p = (k0 >> 1) + (lane >> 4);   // K-pair row in s_Wp
#pragma unroll
        for (int t = 0; t < 4; ++t) {
            const v2f b = *(const v2f*)&s_Wp[p * WP_STRIDE + 2 * (t * 16 + m)];
            acc[t] = __builtin_amdgcn_wmma_f32_16x16x4_f32(
                false, a, false, b, (short)0, acc[t], false, false);
        }
    }

    // D layout: VGPR g, lanes0-15 -> M=g, N=lane; lanes16-31 -> M=8+g, N=lane-16.
    // Spill D[16x64] into the lower half of the a_input tile (item copy done with).
    {
        const int rbase = (lane < 16) ? 0 : 8;
        const int col   = lane & 15;
#pragma unroll
        for (int t = 0; t < 4; ++t) {
#pragma unroll
            for (int g = 0; g < 8; ++g)
                s_a[wave][(rbase + g) * A_STRIDE + t * 16 + col] = acc[t][g];
        }
    }
    __syncthreads();

    // ---- attention logits: lane L handles row k=L&15, half h=L>>4 (32 dims) ----
    const int krow = lane & 15;
    const int h    = lane >> 4;
    const int rid  = kg_relations[item * K_NBR + krow];
    const float* rrow = relation_emb + (size_t)rid * D_EMB;
    float part = 0.0f;
    {
        const int base = krow * A_STRIDE + h * 32;
#pragma unroll
        for (int j = 0; j < 16; ++j) {
            const v2f ov = *(const v2f*)&s_a[wave][base + 2 * j];
            const v2f rv = *(const v2f*)&rrow[h * 32 + 2 * j];
            const int n  = h * 32 + 2 * j;
            part += (ov.x + s_b[n]) * rv.x + (ov.y + s_b[n + 1]) * rv.y;
        }
    }
    float e = part + __shfl_xor(part, 16, 32);            // both halves hold e_k

    e = (e > 0.0f) ? e : 0.2f * e;                        // leaky_relu
    if (kg_entities[item * K_NBR + krow] == PAD_ENT) e = NEG_INF_F;

    // masked softmax over 16 rows (reduction within each 16-lane half; halves identical)
    float mx = e;
#pragma unroll
    for (int s = 1; s < 16; s <<= 1) mx = fmaxf(mx, __shfl_xor(mx, s, 32));
    const float ex = __expf(e - mx);
    float sm = ex;
#pragma unroll
    for (int s = 1; s < 16; s <<= 1) sm += __shfl_xor(sm, s, 32);
    if (lane < 16) s_att[wave][lane] = ex / sm;
    __syncthreads();

    // ---- items_kg = att @ ent + item_emb ; each lane covers d=lane, lane+32 ----
    float o0 = i0, o1 = i1;
    for (int k = 0; k < K_NBR; ++k) {
        const float a = s_att[wave][k];
        o0 += a * s_a[wave][k * A_STRIDE + 64 + lane];
        o1 += a * s_a[wave][k * A_STRIDE + 96 + lane];
    }
    out_items[(size_t)item * D_EMB + lane]      = o0;
    out_items[(size_t)item * D_EMB + lane + 32] = o1;
}

// ---------------------------------------------------------------------------
// Kernel 2: one SpMM layer. One wave per edge, lanes cover D=64 (2 floats/lane).
// Coalesced gather from L2-resident table, HW float atomics scatter.
// ---------------------------------------------------------------------------
__global__ __launch_bounds__(256)
void kgcl_spmm(const int*   __restrict__ g_rows,
               const int*   __restrict__ g_cols,
               const float* __restrict__ g_vals,
               const float* __restrict__ in,
               float*       __restrict__ out,
               int n_edges)
{
    const int wid  = (blockIdx.x * blockDim.x + threadIdx.x) >> 5;
    const int lane = threadIdx.x & 31;
    if (wid >= n_edges) return;
    const int   r = g_rows[wid];
    const int   c = g_cols[wid];
    const float v = g_vals[wid];
    const float* src = in + (size_t)c * D_EMB;
    float* dst = out + (size_t)r * D_EMB;
    unsafeAtomicAdd(&dst[lane],      v * src[lane]);
    unsafeAtomicAdd(&dst[lane + 32], v * src[lane + 32]);
}

// ---------------------------------------------------------------------------
// Kernel 3: acc += x (elementwise)
// ---------------------------------------------------------------------------
__global__ __launch_bounds__(256)
void kgcl_acc_add(float* __restrict__ acc, const float* __restrict__ x, int n)
{
    const int i = blockIdx.x * blockDim.x + threadIdx.x;
    if (i < n) acc[i] += x[i];
}

// ---------------------------------------------------------------------------
// Kernel 4: gamma[b] = dot(acc[users[b]], acc[U+items[b]]) / 16  (1/4 * 1/4)
// ---------------------------------------------------------------------------
__global__ __launch_bounds__(256)
void kgcl_gamma(const int* __restrict__ users,
                const int* __restrict__ items,
                const float* __restrict__ acc,
                float* __restrict__ out,
                int batch, int n_users)
{
    const int wid  = (blockIdx.x * blockDim.x + threadIdx.x) >> 5;
    const int lane = threadIdx.x & 31;
    if (wid >= batch) return;
    const float* u  = acc + (size_t)users[wid] * D_EMB;
    const float* it = acc + (size_t)(n_users + items[wid]) * D_EMB;
    float s = u[lane] * it[lane] + u[lane + 32] * it[lane + 32];
#pragma unroll
    for (int mdel = 16; mdel >= 1; mdel >>= 1) s += __shfl_xor(s, mdel, 32);
    if (lane == 0) out[wid] = s * (1.0f / 16.0f);
}

// ---------------------------------------------------------------------------
extern "C" void kernel_launch(void* const* d_in, const int* in_sizes, int n_in,
                              void* d_out, int out_size, void* d_ws, size_t ws_size,
                              hipStream_t stream)
{
    const int*   users        = (const int*)  d_in[0];
    const int*   items        = (const int*)  d_in[1];
    const int*   g_rows       = (const int*)  d_in[2];
    const int*   g_cols       = (const int*)  d_in[3];
    const float* g_vals       = (const float*)d_in[4];
    const float* user_emb     = (const float*)d_in[5];
    const float* item_emb     = (const float*)d_in[6];
    const float* entity_emb   = (const float*)d_in[7];
    const float* relation_emb = (const float*)d_in[8];
    const int*   kg_entities  = (const int*)  d_in[9];
    const int*   kg_relations = (const int*)  d_in[10];
    const float* fc_W         = (const float*)d_in[11];
    const float* fc_b         = (const float*)d_in[12];
    float*       out          = (float*)d_out;

    const int batch   = in_sizes[0];
    const int n_edges = in_sizes[2];
    const int n_users = in_sizes[5] / D_EMB;      // 50000
    const int n_items = in_sizes[6] / D_EMB;      // 25000
    const int n_nodes = n_users + n_items;        // 75000
    const size_t nd   = (size_t)n_nodes * D_EMB;  // 4.8M floats

    float* buf0 = (float*)d_ws;        // all_emb (layer input)
    float* buf1 = buf0 + nd;           // layer output
    float* acc  = buf1 + nd;           // running sum over layers

    // 1) GAT -> item part of all_emb
    kgcl_gat_wmma<<<n_items / WAVES_PB, WAVES_PB * 32, 0, stream>>>(
        item_emb, entity_emb, relation_emb, kg_entities, kg_relations,
        fc_W, fc_b, buf0 + (size_t)n_users * D_EMB);

    // 2) user part of all_emb, acc = all_emb (layer 0 term)
    hipMemcpyAsync(buf0, user_emb, (size_t)n_users * D_EMB * sizeof(float),
                   hipMemcpyDeviceToDevice, stream);
    hipMemcpyAsync(acc, buf0, nd * sizeof(float), hipMemcpyDeviceToDevice, stream);

    // 3) three propagation layers: buf_next = G @ buf_cur ; acc += buf_next
    float* cur = buf0;
    float* nxt = buf1;
    const int spmm_blocks = (int)(((long long)n_edges * 32 + 255) / 256);
    const int add_blocks  = (int)((nd + 255) / 256);
    for (int l = 0; l < 3; ++l) {
        hipMemsetAsync(nxt, 0, nd * sizeof(float), stream);
        kgcl_spmm<<<spmm_blocks, 256, 0, stream>>>(g_rows, g_cols, g_vals,
                                                   cur, nxt, n_edges);
        kgcl_acc_add<<<add_blocks, 256, 0, stream>>>(acc, nxt, (int)nd);
        float* t = cur; cur = nxt; nxt = t;
    }

    // 4) gamma = <light_u, light_i>, with the 1/(N_LAYERS+1)^2 scale folded in
    const int gamma_blocks = (batch * 32 + 255) / 256;
    kgcl_gamma<<<gamma_blocks, 256, 0, stream>>>(users, items, acc, out,
                                                 batch, n_users);
}